// InfoNCELoss_78829829751155
// MI455X (gfx1250) — compile-verified
//
#include <hip/hip_runtime.h>
#include <hip/hip_bf16.h>
#include <math.h>

typedef float v2f __attribute__((ext_vector_type(2)));
typedef float v8f __attribute__((ext_vector_type(8)));

#define BDIM   32
#define TDIM   256
#define EDIM   256
#define CDIM   256
#define S0     192            // T - MIN_CONTEXT_SIZE
#define ROWS   (BDIM * S0)    // 6144
#define STEPS  12
#define NTOT   71616          // sum_s B*(S0 - s)
#define NEGBIG (-3.0e38f)

// workspace layout (floats)
#define WS_P     0
#define WS_STATS (ROWS * CDIM)                 // 1,572,864
#define WS_POSP  (WS_STATS + ROWS * 16 * 24)   // +2,359,296
#define WS_LSEP  (WS_POSP + 6144)
// total = 3,944,448 floats (~15.8 MB)

__device__ __forceinline__ void lse_merge(float& M, float& S, float m2, float s2) {
    float nm = fmaxf(M, m2);
    S = S * __expf(M - nm) + s2 * __expf(m2 - nm);
    M = nm;
}

// ---------------- Kernel 1: P = emb[:,64:256,:] @ W  (6144x256) ----------------
__global__ __launch_bounds__(32) void proj_kernel(const float* __restrict__ emb,
                                                  const float* __restrict__ W,
                                                  float* __restrict__ P) {
    const int lane = threadIdx.x & 31;
    const int hi   = lane >> 4;     // half-wave
    const int N    = lane & 15;
    const int rowBase = blockIdx.x * 16;
    const int colBase = blockIdx.y * 16;
    const int r = rowBase + N;                 // A-matrix M index = lane&15
    const int b = r / S0, t = r % S0;
    const float* ap = emb + (size_t)(b * TDIM + 64 + t) * EDIM + 2 * hi;
    const float* wp = W + (size_t)(2 * hi) * CDIM + colBase + N;

    v8f acc = {0.f, 0.f, 0.f, 0.f, 0.f, 0.f, 0.f, 0.f};
#pragma unroll
    for (int kc = 0; kc < 64; ++kc) {
        v2f a = *(const v2f*)(ap + 4 * kc);          // K = 4kc + 2hi, +1
        v2f bb;
        bb.x = wp[kc * 4 * CDIM];                    // W[K][col]
        bb.y = wp[kc * 4 * CDIM + CDIM];             // W[K+1][col]
        acc = __builtin_amdgcn_wmma_f32_16x16x4_f32(false, a, false, bb,
                                                    (short)0, acc, false, false);
    }
#pragma unroll
    for (int v = 0; v < 8; ++v)
        P[(size_t)(rowBase + v + 8 * hi) * CDIM + colBase + N] = acc[v];
}

// per-16x16-tile epilogue: per-lane online-softmax updates only (no shuffles)
__device__ __forceinline__ void process_tile(const v8f& acc, int u, int j,
                                             int bRow, int tBase, int hi, int N,
                                             float* mainM, float* mainS,
                                             float* bm, float* bs, float& posAcc) {
    const bool inMain = (u <= 180);
#pragma unroll
    for (int v = 0; v < 8; ++v) {
        float s = acc[v];
        int t = tBase + v + 8 * hi;
        int d = t - u;
        if (j == bRow && d >= 0 && d < STEPS) posAcc += s;   // positive pair
        float sm = inMain ? s : NEGBIG;                      // masked -> no-op merge
        lse_merge(mainM[v], mainS[v], sm, 1.0f);
    }
    if (!inMain) {                       // u in [181,191]: this lane owns bucket u-181
#pragma unroll
        for (int v = 0; v < 8; ++v) lse_merge(bm[v], bs[v], acc[v], 1.0f);
    }
}

// ------- Kernel 2: fused G-tiles + online softmax stats + positives -------
// grid (96 row-blocks, 16 col-chunks), 128 threads = 4 waves; wave owns 16 rows.
// Dual column-tile pipeline: two independent WMMA accumulator chains.
__global__ __launch_bounds__(128) void score_kernel(const float* __restrict__ ctxs,
                                                    const float* __restrict__ P,
                                                    float* __restrict__ stats,
                                                    float* __restrict__ posPartial) {
    const int lane = threadIdx.x & 31;
    const int w    = threadIdx.x >> 5;
    const int hi   = lane >> 4;
    const int N    = lane & 15;
    const int rowTile = blockIdx.x * 64 + w * 16;
    const int chunk   = blockIdx.y;            // 0..15, 384 columns each
    const int colBase = chunk * 384;
    const int bRow  = rowTile / S0;            // row tiles never cross b (192%16==0)
    const int tBase = rowTile % S0;

    // Resident A operands: 16 rows x K=256 -> 64 x v2f = 128 VGPRs
    v2f a[64];
    {
        const float* pp = P + (size_t)(rowTile + N) * CDIM + 2 * hi;
#pragma unroll
        for (int kc = 0; kc < 64; ++kc) a[kc] = *(const v2f*)(pp + 4 * kc);
    }

    // per-LANE accumulators over this lane's own columns
    float mainM[8], mainS[8], bm[8], bs[8];
#pragma unroll
    for (int v = 0; v < 8; ++v) { mainM[v] = NEGBIG; mainS[v] = 0.f; bm[v] = NEGBIG; bs[v] = 0.f; }
    float posAcc = 0.f;

    for (int ct = 0; ct < 24; ct += 2) {
        const int col16a = colBase + ct * 16;
        const int col16b = col16a + 16;
        const int cola = col16a + N,  colb = col16b + N;
        const int ja = cola / S0,     jb = colb / S0;
        const int ua = cola % S0,     ub = colb % S0;
        const float* cpa = ctxs + (size_t)(ja * TDIM + 63 + ua) * CDIM + 2 * hi;
        const float* cpb = ctxs + (size_t)(jb * TDIM + 63 + ub) * CDIM + 2 * hi;

        if (ct + 2 < 24) {   // prefetch next tile-pair's context rows (global_prefetch_b8)
            const int colc = col16b + 16 + N;
            const float* cpn = ctxs + (size_t)((colc / S0) * TDIM + 63 + (colc % S0)) * CDIM;
            __builtin_prefetch(cpn, 0, 1);
        }

        v8f acc0 = {0.f, 0.f, 0.f, 0.f, 0.f, 0.f, 0.f, 0.f};
        v8f acc1 = {0.f, 0.f, 0.f, 0.f, 0.f, 0.f, 0.f, 0.f};
#pragma unroll
        for (int kc = 0; kc < 64; ++kc) {
            v2f b0 = *(const v2f*)(cpa + 4 * kc);
            v2f b1 = *(const v2f*)(cpb + 4 * kc);
            acc0 = __builtin_amdgcn_wmma_f32_16x16x4_f32(false, a[kc], false, b0,
                                                         (short)0, acc0, false, false);
            acc1 = __builtin_amdgcn_wmma_f32_16x16x4_f32(false, a[kc], false, b1,
                                                         (short)0, acc1, false, false);
        }
        process_tile(acc0, ua, ja, bRow, tBase, hi, N, mainM, mainS, bm, bs, posAcc);
        process_tile(acc1, ub, jb, bRow, tBase, hi, N, mainM, mainS, bm, bs, posAcc);
    }

    // ONE cross-lane butterfly merge per row-half (lanes sharing hi)
#pragma unroll
    for (int v = 0; v < 8; ++v) {
#pragma unroll
        for (int m = 1; m < 16; m <<= 1) {
            float oM = __shfl_xor(mainM[v], m, 32);
            float oS = __shfl_xor(mainS[v], m, 32);
            lse_merge(mainM[v], mainS[v], oM, oS);
        }
    }

    // store per-(row,chunk) stats: [0,1]=main (m,S); [2+2k,3+2k]=tail u=181+k
    if (N == 0) {
#pragma unroll
        for (int v = 0; v < 8; ++v) {
            size_t idx = ((size_t)(rowTile + v + 8 * hi) * 16 + chunk) * 24;
            stats[idx + 0] = mainM[v];
            stats[idx + 1] = mainS[v];
        }
    }
    if (N >= 5) {
#pragma unroll
        for (int v = 0; v < 8; ++v) {
            size_t idx = ((size_t)(rowTile + v + 8 * hi) * 16 + chunk) * 24 + 2 + 2 * (N - 5);
            stats[idx + 0] = bm[v];
            stats[idx + 1] = bs[v];
        }
    }
    // per-wave positives partial (deterministic; no float atomics)
#pragma unroll
    for (int m = 1; m < 32; m <<= 1) posAcc += __shfl_xor(posAcc, m, 32);
    if (lane == 0) {
        int wid = (blockIdx.y * gridDim.x + blockIdx.x) * 4 + w;
        posPartial[wid] = posAcc;
    }
}

// -------- Kernel 3: per-row combine chunks + 12 suffix logsumexps --------
__global__ __launch_bounds__(128) void finalize_rows(const float* __restrict__ stats,
                                                     float* __restrict__ lsePartial) {
    int r = blockIdx.x * blockDim.x + threadIdx.x;
    if (r >= ROWS) return;
    int t = r % S0;
    const float* st = stats + (size_t)r * 16 * 24;
    float M = st[0], S = st[1];
    float bm[11], bs[11];
#pragma unroll
    for (int k = 0; k < 11; ++k) { bm[k] = st[2 + 2 * k]; bs[k] = st[3 + 2 * k]; }
    for (int c = 1; c < 16; ++c) {
        const float* sc = st + c * 24;
        lse_merge(M, S, sc[0], sc[1]);
#pragma unroll
        for (int k = 0; k < 11; ++k) lse_merge(bm[k], bs[k], sc[2 + 2 * k], sc[3 + 2 * k]);
    }
    // step s includes columns u < 192 - s : walk s = 11 .. 0, adding tail u = 192-s
    float total = 0.f;
    for (int s = STEPS - 1; s >= 0; --s) {
        if (s <= t) total += (M + logf(S)) - logf(32.0f * (float)(S0 - s));
        if (s > 0) lse_merge(M, S, bm[11 - s], bs[11 - s]);
    }
    lsePartial[r] = total;
}

// -------- Kernel 4: deterministic final reduction --------
__global__ __launch_bounds__(256) void reduce_kernel(const float* __restrict__ posPartial,
                                                     const float* __restrict__ lsePartial,
                                                     float* __restrict__ out) {
    __shared__ float sh[256];
    float acc = 0.f;
    for (int i = threadIdx.x; i < 6144; i += 256)
        acc += lsePartial[i] - posPartial[i];
    sh[threadIdx.x] = acc;
    __syncthreads();
    for (int s = 128; s > 0; s >>= 1) {
        if (threadIdx.x < s) sh[threadIdx.x] += sh[threadIdx.x + s];
        __syncthreads();
    }
    if (threadIdx.x == 0) out[0] = sh[0] / (float)NTOT;
}

extern "C" void kernel_launch(void* const* d_in, const int* in_sizes, int n_in,
                              void* d_out, int out_size, void* d_ws, size_t ws_size,
                              hipStream_t stream) {
    (void)in_sizes; (void)n_in; (void)out_size; (void)ws_size;
    const float* emb = (const float*)d_in[0];   // (32,256,256)
    const float* ctx = (const float*)d_in[1];   // (32,256,256)
    const float* W   = (const float*)d_in[2];   // (256,256)
    float* ws    = (float*)d_ws;
    float* P     = ws + WS_P;
    float* stats = ws + WS_STATS;
    float* posP  = ws + WS_POSP;
    float* lseP  = ws + WS_LSEP;

    proj_kernel<<<dim3(ROWS / 16, CDIM / 16), 32, 0, stream>>>(emb, W, P);
    score_kernel<<<dim3(ROWS / 64, 16), 128, 0, stream>>>(ctx, P, stats, posP);
    finalize_rows<<<dim3(ROWS / 128), 128, 0, stream>>>(stats, lseP);
    reduce_kernel<<<1, 256, 0, stream>>>(posP, lseP, (float*)d_out);
}